// KGAT_53206054863053
// MI455X (gfx1250) — compile-verified
//
#include <hip/hip_runtime.h>
#include <hip/hip_bf16.h>
#include <math.h>

// ---------------------------------------------------------------------------
// KGAT forward for MI455X (gfx1250, wave32).
// Matrix work (TransR projection + bi-interaction MLPs) runs on
// V_WMMA_F32_16X16X4_F32 so fp32 semantics match the JAX reference exactly.
// ---------------------------------------------------------------------------

typedef __attribute__((ext_vector_type(2))) float v2f;
typedef __attribute__((ext_vector_type(8))) float v8f;

#define N_ENT 50000
#define N_REL 16
#define DIM 64

// ---------------------------------------------------------------------------
// Kernel 1: proj[r] = ent_embed @ W_R[r]   ([N,64] x [64,64] -> [R,N,64])
// Block = 128 threads (4 waves). Each block: one 16-row tile, one relation.
// Each wave owns one 16-col tile; K=64 swept in 16 WMMA steps of K=4.
// ---------------------------------------------------------------------------
__global__ __launch_bounds__(128) void proj_gemm_kernel(
    const float* __restrict__ ent, const float* __restrict__ W_R,
    float* __restrict__ proj, int N)
{
    __shared__ float As[16][DIM];   // 4 KB
    __shared__ float Bs[DIM][DIM];  // 16 KB
    const int r = blockIdx.y;
    const int rowBase = blockIdx.x * 16;
    const int tid = threadIdx.x;

    // Stage A tile (16x64) and W_R[r] (64x64) into LDS, float4 vectorized.
    const float4* Ag = (const float4*)(ent + (size_t)rowBase * DIM);
    float4* Al = (float4*)&As[0][0];
    for (int i = tid; i < 16 * DIM / 4; i += 128) Al[i] = Ag[i];
    const float4* Bg = (const float4*)(W_R + (size_t)r * DIM * DIM);
    float4* Bl = (float4*)&Bs[0][0];
    for (int i = tid; i < DIM * DIM / 4; i += 128) Bl[i] = Bg[i];
    __syncthreads();

    const int wave = tid >> 5;
    const int lane = tid & 31;
    const int row  = lane & 15;     // A-fragment: M = lane%16
    const int half = lane >> 4;     // K phase: 0 -> K{0,1}, 1 -> K{2,3}
    const int col  = (lane & 15) + wave * 16;

    v8f acc = {};
    #pragma unroll
    for (int t = 0; t < DIM / 4; ++t) {
        const int k = t * 4 + half * 2;
        v2f a = *(const v2f*)&As[row][k];          // contiguous pair, 8B aligned
        v2f b; b.x = Bs[k][col]; b.y = Bs[k + 1][col];
        acc = __builtin_amdgcn_wmma_f32_16x16x4_f32(
            false, a, false, b, (short)0, acc, false, false);
    }

    float* out = proj + (size_t)r * N * DIM;
    #pragma unroll
    for (int v = 0; v < 8; ++v) {
        const int rr = v + 8 * half;               // C/D layout: M = v + 8*half
        out[(size_t)(rowBase + rr) * DIM + col] = acc[v];
    }
}

// ---------------------------------------------------------------------------
// Kernel 2: per-edge attention logit. One wave32 per edge; lane holds a
// float2 slice of the 64-dim projected rows (coalesced 256B row reads).
// ---------------------------------------------------------------------------
__global__ __launch_bounds__(256) void att_kernel(
    const float* __restrict__ proj, const float* __restrict__ rel_embed,
    const int* __restrict__ src, const int* __restrict__ dst,
    const int* __restrict__ etype, float* __restrict__ att, int E, int N)
{
    const int e = blockIdx.x * 8 + (threadIdx.x >> 5);
    if (e >= E) return;
    const int lane = threadIdx.x & 31;
    const int r = etype[e], s = src[e], d = dst[e];
    const float2 t  = ((const float2*)(proj + ((size_t)r * N + s) * DIM))[lane];
    const float2 h  = ((const float2*)(proj + ((size_t)r * N + d) * DIM))[lane];
    const float2 re = ((const float2*)(rel_embed + (size_t)r * DIM))[lane];
    float p = t.x * tanhf(h.x + re.x) + t.y * tanhf(h.y + re.y);
    #pragma unroll
    for (int off = 16; off > 0; off >>= 1) p += __shfl_xor(p, off);
    if (lane == 0) att[e] = p;
}

// Monotone float <-> uint mapping for atomic segment-max (init value 0u == min).
__device__ __forceinline__ unsigned f2ord(float f) {
    unsigned b = __float_as_uint(f);
    return (b & 0x80000000u) ? ~b : (b | 0x80000000u);
}
__device__ __forceinline__ float ord2f(unsigned u) {
    return __uint_as_float((u & 0x80000000u) ? (u & 0x7FFFFFFFu) : ~u);
}

__global__ void segmax_kernel(const float* __restrict__ att,
                              const int* __restrict__ dst,
                              unsigned* __restrict__ nmax, int E)
{
    const int e = blockIdx.x * blockDim.x + threadIdx.x;
    if (e >= E) return;
    atomicMax(&nmax[dst[e]], f2ord(att[e]));
}

__global__ void expsum_kernel(float* __restrict__ att,
                              const int* __restrict__ dst,
                              const unsigned* __restrict__ nmax,
                              float* __restrict__ nsum, int E)
{
    const int e = blockIdx.x * blockDim.x + threadIdx.x;
    if (e >= E) return;
    const int d = dst[e];
    const float ev = __expf(att[e] - ord2f(nmax[d]));
    att[e] = ev;                         // att becomes unnormalized weight
    atomicAdd(&nsum[d], ev);
}

__global__ void adiv_kernel(float* __restrict__ a, const int* __restrict__ dst,
                            const float* __restrict__ nsum, int E)
{
    const int e = blockIdx.x * blockDim.x + threadIdx.x;
    if (e >= E) return;
    a[e] = a[e] / nsum[dst[e]];
}

// ---------------------------------------------------------------------------
// Kernel 3: message scatter  N_h[dst] += a[e] * h[src]  (per edge-feature)
// ---------------------------------------------------------------------------
template <int DIN>
__global__ void scatter_kernel(const float* __restrict__ h,
                               const float* __restrict__ a,
                               const int* __restrict__ src,
                               const int* __restrict__ dst,
                               float* __restrict__ Nh, int E)
{
    const long long idx = (long long)blockIdx.x * blockDim.x + threadIdx.x;
    if (idx >= (long long)E * DIN) return;
    const int e = (int)(idx / DIN);
    const int k = (int)(idx % DIN);
    atomicAdd(&Nh[(size_t)dst[e] * DIN + k],
              a[e] * h[(size_t)src[e] * DIN + k]);
}

// ---------------------------------------------------------------------------
// Kernel 4: bi-interaction MLP (WMMA).
//   out = lrelu((h+Nh)@W1 + b1) + lrelu((h*Nh)@W2 + b2)
// Block = DOUT/16 waves; 16 nodes per block; wave per 16-col tile.
// ---------------------------------------------------------------------------
template <int DIN, int DOUT>
__global__ __launch_bounds__(DOUT * 2) void bi_mlp_kernel(
    const float* __restrict__ h, const float* __restrict__ Nh,
    const float* __restrict__ W1, const float* __restrict__ b1,
    const float* __restrict__ W2, const float* __restrict__ b2,
    float* __restrict__ hout)
{
    __shared__ float Ss[16][DIN];
    __shared__ float Ps[16][DIN];
    __shared__ float W1s[DIN][DOUT];
    __shared__ float W2s[DIN][DOUT];
    const int nthreads = (DOUT / 16) * 32;
    const int tid = threadIdx.x;
    const int nodeBase = blockIdx.x * 16;

    for (int i = tid; i < 16 * DIN; i += nthreads) {
        const int n = i / DIN, k = i % DIN;
        const float hv = h[(size_t)(nodeBase + n) * DIN + k];
        const float nv = Nh[(size_t)(nodeBase + n) * DIN + k];
        Ss[n][k] = hv + nv;
        Ps[n][k] = hv * nv;
    }
    for (int i = tid; i < DIN * DOUT; i += nthreads) {
        W1s[i / DOUT][i % DOUT] = W1[i];
        W2s[i / DOUT][i % DOUT] = W2[i];
    }
    __syncthreads();

    const int wave = tid >> 5;
    const int lane = tid & 31;
    const int row  = lane & 15;
    const int half = lane >> 4;
    const int col  = (lane & 15) + wave * 16;

    v8f acc1 = {}, acc2 = {};
    #pragma unroll
    for (int t = 0; t < DIN / 4; ++t) {
        const int k = t * 4 + half * 2;
        v2f a1 = *(const v2f*)&Ss[row][k];
        v2f b1v; b1v.x = W1s[k][col]; b1v.y = W1s[k + 1][col];
        acc1 = __builtin_amdgcn_wmma_f32_16x16x4_f32(
            false, a1, false, b1v, (short)0, acc1, false, false);
        v2f a2 = *(const v2f*)&Ps[row][k];
        v2f b2v; b2v.x = W2s[k][col]; b2v.y = W2s[k + 1][col];
        acc2 = __builtin_amdgcn_wmma_f32_16x16x4_f32(
            false, a2, false, b2v, (short)0, acc2, false, false);
    }

    const float bb1 = b1[col], bb2 = b2[col];
    #pragma unroll
    for (int v = 0; v < 8; ++v) {
        const int rr = v + 8 * half;
        float x1 = acc1[v] + bb1;
        float x2 = acc2[v] + bb2;
        x1 = x1 > 0.0f ? x1 : 0.01f * x1;
        x2 = x2 > 0.0f ? x2 : 0.01f * x2;
        hout[(size_t)(nodeBase + rr) * DOUT + col] = x1 + x2;
    }
}

// ---------------------------------------------------------------------------
// Kernel 5: l2-normalize a layer output into the [N,176] concat buffer.
// ---------------------------------------------------------------------------
template <int DOUT>
__global__ void norm_out_kernel(const float* __restrict__ hout,
                                float* __restrict__ out, int N, int off)
{
    const int n = blockIdx.x * blockDim.x + threadIdx.x;
    if (n >= N) return;
    float vals[DOUT];
    float ss = 0.0f;
    #pragma unroll
    for (int k = 0; k < DOUT; ++k) {
        const float v = hout[(size_t)n * DOUT + k];
        vals[k] = v;
        ss += v * v;
    }
    const float inv = 1.0f / fmaxf(sqrtf(ss), 1e-12f);
    #pragma unroll
    for (int k = 0; k < DOUT; ++k)
        out[(size_t)n * 176 + off + k] = vals[k] * inv;
}

__global__ void copy_h0_kernel(const float* __restrict__ ent,
                               float* __restrict__ out, int N)
{
    const int idx = blockIdx.x * blockDim.x + threadIdx.x;
    if (idx >= N * DIM) return;
    const int n = idx / DIM, k = idx % DIM;
    out[(size_t)n * 176 + k] = ent[idx];
}

// ---------------------------------------------------------------------------
extern "C" void kernel_launch(void* const* d_in, const int* in_sizes, int n_in,
                              void* d_out, int out_size, void* d_ws, size_t ws_size,
                              hipStream_t stream)
{
    const float* ent   = (const float*)d_in[0];
    const float* rel   = (const float*)d_in[1];
    const float* W_R   = (const float*)d_in[2];
    const float* W1_0  = (const float*)d_in[3];
    const float* b1_0  = (const float*)d_in[4];
    const float* W2_0  = (const float*)d_in[5];
    const float* b2_0  = (const float*)d_in[6];
    const float* W1_1  = (const float*)d_in[7];
    const float* b1_1  = (const float*)d_in[8];
    const float* W2_1  = (const float*)d_in[9];
    const float* b2_1  = (const float*)d_in[10];
    const float* W1_2  = (const float*)d_in[11];
    const float* b1_2  = (const float*)d_in[12];
    const float* W2_2  = (const float*)d_in[13];
    const float* b2_2  = (const float*)d_in[14];
    const int*   src   = (const int*)d_in[15];
    const int*   dst   = (const int*)d_in[16];
    const int*   etype = (const int*)d_in[17];

    const int N = N_ENT;
    const int E = in_sizes[15];
    float* out = (float*)d_out;

    // Workspace carve-up (floats).
    float* ws = (float*)d_ws;
    size_t off = 0;
    float*    proj = ws + off;            off += (size_t)N_REL * N * DIM; // 204.8 MB
    float*    att  = ws + off;            off += (size_t)E;               // logits -> a
    unsigned* nmax = (unsigned*)(ws + off); off += (size_t)N;
    float*    nsum = ws + off;            off += (size_t)N;
    float*    Nh   = ws + off;            off += (size_t)N * DIM;
    float*    h1   = ws + off;            off += (size_t)N * DIM;
    float*    h2   = ws + off;            off += (size_t)N * 32;
    float*    h3   = ws + off;            off += (size_t)N * 16;

    // --- Phase 1: TransR projections (WMMA GEMMs) ---
    proj_gemm_kernel<<<dim3(N / 16, N_REL), 128, 0, stream>>>(ent, W_R, proj, N);

    // --- Phase 2: attention logits + edge softmax grouped by dst ---
    att_kernel<<<(E + 7) / 8, 256, 0, stream>>>(proj, rel, src, dst, etype, att, E, N);
    hipMemsetAsync(nmax, 0, (size_t)N * sizeof(unsigned), stream); // ordered-min
    hipMemsetAsync(nsum, 0, (size_t)N * sizeof(float), stream);
    segmax_kernel<<<(E + 255) / 256, 256, 0, stream>>>(att, dst, nmax, E);
    expsum_kernel<<<(E + 255) / 256, 256, 0, stream>>>(att, dst, nmax, nsum, E);
    adiv_kernel<<<(E + 255) / 256, 256, 0, stream>>>(att, dst, nsum, E);

    // --- Phase 3: three bi-interaction layers ---
    // Layer 0: 64 -> 64  (input h = ent_embed)
    hipMemsetAsync(Nh, 0, (size_t)N * 64 * sizeof(float), stream);
    {
        long long tot = (long long)E * 64;
        scatter_kernel<64><<<(unsigned)((tot + 255) / 256), 256, 0, stream>>>(
            ent, att, src, dst, Nh, E);
    }
    bi_mlp_kernel<64, 64><<<N / 16, 128, 0, stream>>>(ent, Nh, W1_0, b1_0, W2_0, b2_0, h1);
    norm_out_kernel<64><<<(N + 127) / 128, 128, 0, stream>>>(h1, out, N, 64);

    // Layer 1: 64 -> 32  (input h = h1, un-normalized)
    hipMemsetAsync(Nh, 0, (size_t)N * 64 * sizeof(float), stream);
    {
        long long tot = (long long)E * 64;
        scatter_kernel<64><<<(unsigned)((tot + 255) / 256), 256, 0, stream>>>(
            h1, att, src, dst, Nh, E);
    }
    bi_mlp_kernel<64, 32><<<N / 16, 64, 0, stream>>>(h1, Nh, W1_1, b1_1, W2_1, b2_1, h2);
    norm_out_kernel<32><<<(N + 127) / 128, 128, 0, stream>>>(h2, out, N, 128);

    // Layer 2: 32 -> 16  (input h = h2)
    hipMemsetAsync(Nh, 0, (size_t)N * 32 * sizeof(float), stream);
    {
        long long tot = (long long)E * 32;
        scatter_kernel<32><<<(unsigned)((tot + 255) / 256), 256, 0, stream>>>(
            h2, att, src, dst, Nh, E);
    }
    bi_mlp_kernel<32, 16><<<N / 16, 32, 0, stream>>>(h2, Nh, W1_2, b1_2, W2_2, b2_2, h3);
    norm_out_kernel<16><<<(N + 127) / 128, 128, 0, stream>>>(h3, out, N, 160);

    // Slot 0: raw ent_embed (first element of all_embed is NOT normalized)
    copy_h0_kernel<<<(N * DIM + 255) / 256, 256, 0, stream>>>(ent, out, N);
}